// SD2D_68375879352765
// MI455X (gfx1250) — compile-verified
//
#include <hip/hip_runtime.h>
#include <hip/hip_bf16.h>

// ---------------------------------------------------------------------------
// Shapes: B=2, C=256, W=H=512; half-res plane n = 256*256 = 65536.
// ws layout: avg [B*C*65536] floats (128 MB)
//            Wp  [65536] floats  -- (W2@W1) packed as K-pairs:
//                 Wp[((k>>1)*256 + o)*2 + (k&1)] = Wc[o,k]
//            ssq [B*C*4] floats (sum of squares of raw basis)
// origin is stored in d_out (same element count), overwritten by pass2.
// ---------------------------------------------------------------------------

typedef float v2f __attribute__((ext_vector_type(2)));
typedef float v8f __attribute__((ext_vector_type(8)));

#define NPLANE 65536
#define HSTRIDE 512
#define EPSF 1e-12f

// ---------------------------------------------------------------------------
// Kernel 1: Wc = W2@W1 (33 MFLOP), stored K-pair-packed for b64 A-operand loads.
// ---------------------------------------------------------------------------
__global__ __launch_bounds__(256) void compose_wt(const float* __restrict__ W1,
                                                  const float* __restrict__ W2,
                                                  float* __restrict__ Wp) {
    const int c = blockIdx.x;   // 0..255  (K index of Wc)
    const int o = threadIdx.x;  // 0..255  (M index of Wc)
    float s = 0.f;
    #pragma unroll 4
    for (int m = 0; m < 256; ++m)
        s = fmaf(W2[o * 256 + m], W1[m * 256 + c], s);
    Wp[(((c >> 1) * 256 + o) << 1) + (c & 1)] = s;
}

// ---------------------------------------------------------------------------
// Kernel 2: one block per (b,c) plane; b128 streaming of cen.
//   avg[plane][pos] = 0.25*(a+p+q+r)
//   ssq[plane][k]   = sum_n basis_k^2   (k = mx, e1, e2, e3)
// Each thread handles 2 adjacent half-res columns per iteration.
// ---------------------------------------------------------------------------
__global__ __launch_bounds__(256) void pass1(const float* __restrict__ cen,
                                             float* __restrict__ avg,
                                             float* __restrict__ ssq) {
    const int plane = blockIdx.x;                       // b*C + c
    const float* src = cen + (size_t)plane * (HSTRIDE * HSTRIDE);
    float* avgp = avg + (size_t)plane * NPLANE;
    const int tid = threadIdx.x;

    float lss0 = 0.f, lss1 = 0.f, lss2 = 0.f, lss3 = 0.f;

    for (int it = 0; it < 128; ++it) {
        const int pos2 = it * 512 + tid * 2;            // even position
        const int i = pos2 >> 8;
        const int j = pos2 & 255;                       // even column
        const float4 ap4 = *(const float4*)(src + ((size_t)(2 * i) << 9) + 2 * j);
        const float4 qr4 = *(const float4*)(src + ((size_t)(2 * i + 1) << 9) + 2 * j);

        // position 0: a,p,q,r = ap4.x, ap4.y, qr4.x, qr4.y
        // position 1: a,p,q,r = ap4.z, ap4.w, qr4.z, qr4.w
        float2 av;
        av.x = 0.25f * (ap4.x + ap4.y + qr4.x + qr4.y);
        av.y = 0.25f * (ap4.z + ap4.w + qr4.z + qr4.w);
        *(float2*)(avgp + pos2) = av;

        {
            const float mx = fmaxf(fmaxf(ap4.x, ap4.y), fmaxf(qr4.x, qr4.y));
            const float e1 = ap4.x - ap4.y + qr4.x - qr4.y;
            const float e2 = ap4.x + ap4.y - qr4.x - qr4.y;
            const float e3 = ap4.x - ap4.y - qr4.x + qr4.y;
            lss0 = fmaf(mx, mx, lss0);
            lss1 = fmaf(e1, e1, lss1);
            lss2 = fmaf(e2, e2, lss2);
            lss3 = fmaf(e3, e3, lss3);
        }
        {
            const float mx = fmaxf(fmaxf(ap4.z, ap4.w), fmaxf(qr4.z, qr4.w));
            const float e1 = ap4.z - ap4.w + qr4.z - qr4.w;
            const float e2 = ap4.z + ap4.w - qr4.z - qr4.w;
            const float e3 = ap4.z - ap4.w - qr4.z + qr4.w;
            lss0 = fmaf(mx, mx, lss0);
            lss1 = fmaf(e1, e1, lss1);
            lss2 = fmaf(e2, e2, lss2);
            lss3 = fmaf(e3, e3, lss3);
        }
    }

    __shared__ float red[4][256];
    red[0][tid] = lss0; red[1][tid] = lss1; red[2][tid] = lss2; red[3][tid] = lss3;
    __syncthreads();
    for (int s = 128; s > 0; s >>= 1) {
        if (tid < s) {
            #pragma unroll
            for (int k = 0; k < 4; ++k) red[k][tid] += red[k][tid + s];
        }
        __syncthreads();
    }
    if (tid < 4) ssq[plane * 4 + tid] = red[tid][0];
}

// ---------------------------------------------------------------------------
// Kernel 3: origin[b,o,n] = sum_c Wc[o,c]*avg[b,c,n] + b2[o]
// WMMA f32 16x16x4. A-operand: single b64 from K-pair-packed Wp (L2 resident);
// B-operand: avg tile staged in LDS (ds_load_2addr pairs).
// Per ISA layout: A/B vgpr0 holds K={0,2} on lane halves, vgpr1 K={1,3};
// C/D vgpr v holds rows M=v (lanes 0-15) and M=v+8 (lanes 16-31).
// Block: 256 threads = 8 waves; wave w owns o-tiles w*32 and w*32+16; N-tile = 16.
// ---------------------------------------------------------------------------
__global__ __launch_bounds__(256) void gemm_wmma(const float* __restrict__ avg,
                                                 const float* __restrict__ Wp,
                                                 const float* __restrict__ b2,
                                                 float* origin) {
    const int b    = blockIdx.y;
    const int col0 = blockIdx.x * 16;
    const int tid  = threadIdx.x;
    const int wave = tid >> 5;
    const int lane = tid & 31;

    __shared__ float bt[256 * 16];  // [c][col] : 16 KB

    const float* avgb = avg + (size_t)b * ((size_t)256 * NPLANE);

    // Cooperative stage of B tile: thread t loads avg row c=t, 16 cols (64B contig)
    {
        const float4* s4 = (const float4*)(avgb + (size_t)tid * NPLANE + col0);
        float4* dst = (float4*)(bt + tid * 16);
        dst[0] = s4[0]; dst[1] = s4[1]; dst[2] = s4[2]; dst[3] = s4[3];
    }
    __syncthreads();

    const int hsel = lane >> 4;            // 0 / 1 (lane half)
    const int lcol = lane & 15;
    const int o0 = wave * 32;
    const int o1 = o0 + 16;
    const v2f* wp2 = (const v2f*)Wp;       // wp2[kp*256 + o] = {Wc[o,2kp], Wc[o,2kp+1]}

    v8f acc0 = {};
    v8f acc1 = {};

    for (int k = 0; k < 256; k += 4) {
        const int k0 = k + (hsel << 1);    // K base for this lane half (0 or 2 offset)
        const int kp = (k >> 1) + hsel;    // K-pair index
        v2f Bv;
        Bv.x = bt[k0 * 16 + lcol];
        Bv.y = bt[(k0 + 1) * 16 + lcol];
        const v2f A0 = wp2[(size_t)kp * 256 + o0 + lcol];   // global_load_b64
        const v2f A1 = wp2[(size_t)kp * 256 + o1 + lcol];   // global_load_b64

        acc0 = __builtin_amdgcn_wmma_f32_16x16x4_f32(false, A0, false, Bv,
                                                     (short)0, acc0, false, false);
        acc1 = __builtin_amdgcn_wmma_f32_16x16x4_f32(false, A1, false, Bv,
                                                     (short)0, acc1, false, false);
    }

    const int rbase = (lane < 16) ? 0 : 8;
    #pragma unroll
    for (int v = 0; v < 8; ++v) {
        const int r0 = o0 + v + rbase;
        const int r1 = o1 + v + rbase;
        origin[((size_t)b * 256 + r0) * NPLANE + col0 + lcol] = acc0[v] + b2[r0];
        origin[((size_t)b * 256 + r1) * NPLANE + col0 + lcol] = acc1[v] + b2[r1];
    }
}

// ---------------------------------------------------------------------------
// u_k from raw 2x2 block values: scaled basis -> 4x4 ReLU -> 4x4 ReLU
// ---------------------------------------------------------------------------
__device__ __forceinline__ void compute_u_vals(float a, float p, float q, float r,
                                               const float scl[4],
                                               const float m1[16], const float c1[4],
                                               const float m2[16], const float c2[4],
                                               float u[4]) {
    float sb[4];
    sb[0] = fmaxf(fmaxf(a, p), fmaxf(q, r)) * scl[0];
    sb[1] = (a - p + q - r) * scl[1];
    sb[2] = (a + p - q - r) * scl[2];
    sb[3] = (a - p - q + r) * scl[3];

    float h[4];
    #pragma unroll
    for (int jj = 0; jj < 4; ++jj) {
        float s = c1[jj];
        #pragma unroll
        for (int kk = 0; kk < 4; ++kk) s = fmaf(m1[jj * 4 + kk], sb[kk], s);
        h[jj] = fmaxf(s, 0.f);
    }
    #pragma unroll
    for (int jj = 0; jj < 4; ++jj) {
        float s = c2[jj];
        #pragma unroll
        for (int kk = 0; kk < 4; ++kk) s = fmaf(m2[jj * 4 + kk], h[kk], s);
        u[jj] = fmaxf(s, 0.f);
    }
}

// ---------------------------------------------------------------------------
// Kernel 4: one block per plane, b128 streaming, 2 positions/thread/iter.
//   sweep A: dot_k = <origin, u_k>, ss1_k = |u_k|^2  (origin aliases d_out)
//   reduce  : coef_k = dot_k / max(sqrt(ss1_k), eps)^2
//   sweep B: out[pos] = sum_k coef_k * u_k[pos]      (overwrites origin plane)
// ---------------------------------------------------------------------------
__global__ __launch_bounds__(256) void pass2(const float* __restrict__ cen,
                                             const float* origin,
                                             const float* __restrict__ ssq,
                                             const float* __restrict__ ns1_w,
                                             const float* __restrict__ ns1_b,
                                             const float* __restrict__ ns2_w,
                                             const float* __restrict__ ns2_b,
                                             float* out) {
    const int plane = blockIdx.x;
    const float* src  = cen + (size_t)plane * (HSTRIDE * HSTRIDE);
    const float* orgp = origin + (size_t)plane * NPLANE;
    float* outp = out + (size_t)plane * NPLANE;
    const int tid = threadIdx.x;

    float m1[16], m2[16], c1[4], c2[4], scl[4];
    #pragma unroll
    for (int i = 0; i < 16; ++i) { m1[i] = ns1_w[i]; m2[i] = ns2_w[i]; }
    #pragma unroll
    for (int k = 0; k < 4; ++k) {
        c1[k] = ns1_b[k];
        c2[k] = ns2_b[k];
        scl[k] = 0.5f / fmaxf(sqrtf(ssq[plane * 4 + k]), EPSF);
    }

    float dotk[4] = {0.f, 0.f, 0.f, 0.f};
    float ss1[4]  = {0.f, 0.f, 0.f, 0.f};

    for (int it = 0; it < 128; ++it) {
        const int pos2 = it * 512 + tid * 2;
        const int i = pos2 >> 8;
        const int j = pos2 & 255;
        const float4 ap4 = *(const float4*)(src + ((size_t)(2 * i) << 9) + 2 * j);
        const float4 qr4 = *(const float4*)(src + ((size_t)(2 * i + 1) << 9) + 2 * j);
        const float2 og = *(const float2*)(orgp + pos2);

        float u0[4], u1[4];
        compute_u_vals(ap4.x, ap4.y, qr4.x, qr4.y, scl, m1, c1, m2, c2, u0);
        compute_u_vals(ap4.z, ap4.w, qr4.z, qr4.w, scl, m1, c1, m2, c2, u1);
        #pragma unroll
        for (int k = 0; k < 4; ++k) {
            dotk[k] = fmaf(og.x, u0[k], dotk[k]);
            dotk[k] = fmaf(og.y, u1[k], dotk[k]);
            ss1[k]  = fmaf(u0[k], u0[k], ss1[k]);
            ss1[k]  = fmaf(u1[k], u1[k], ss1[k]);
        }
    }

    __shared__ float red[8][256];
    #pragma unroll
    for (int k = 0; k < 4; ++k) { red[k][tid] = dotk[k]; red[k + 4][tid] = ss1[k]; }
    __syncthreads();
    for (int s = 128; s > 0; s >>= 1) {
        if (tid < s) {
            #pragma unroll
            for (int k = 0; k < 8; ++k) red[k][tid] += red[k][tid + s];
        }
        __syncthreads();
    }

    __shared__ float coef[4];
    if (tid < 4) {
        const float nm = fmaxf(sqrtf(red[tid + 4][0]), EPSF);
        coef[tid] = red[tid][0] / (nm * nm);
    }
    __syncthreads();
    const float cf0 = coef[0], cf1 = coef[1], cf2 = coef[2], cf3 = coef[3];

    for (int it = 0; it < 128; ++it) {
        const int pos2 = it * 512 + tid * 2;
        const int i = pos2 >> 8;
        const int j = pos2 & 255;
        const float4 ap4 = *(const float4*)(src + ((size_t)(2 * i) << 9) + 2 * j);
        const float4 qr4 = *(const float4*)(src + ((size_t)(2 * i + 1) << 9) + 2 * j);

        float u0[4], u1[4];
        compute_u_vals(ap4.x, ap4.y, qr4.x, qr4.y, scl, m1, c1, m2, c2, u0);
        compute_u_vals(ap4.z, ap4.w, qr4.z, qr4.w, scl, m1, c1, m2, c2, u1);

        float2 o2;
        o2.x = fmaf(cf0, u0[0], fmaf(cf1, u0[1], fmaf(cf2, u0[2], cf3 * u0[3])));
        o2.y = fmaf(cf0, u1[0], fmaf(cf1, u1[1], fmaf(cf2, u1[2], cf3 * u1[3])));
        *(float2*)(outp + pos2) = o2;
    }
}

// ---------------------------------------------------------------------------
extern "C" void kernel_launch(void* const* d_in, const int* in_sizes, int n_in,
                              void* d_out, int out_size, void* d_ws, size_t ws_size,
                              hipStream_t stream) {
    const float* cen = (const float*)d_in[0];
    const float* W1  = (const float*)d_in[1];
    const float* W2  = (const float*)d_in[2];
    const float* b2  = (const float*)d_in[3];
    const float* n1w = (const float*)d_in[4];
    const float* n1b = (const float*)d_in[5];
    const float* n2w = (const float*)d_in[6];
    const float* n2b = (const float*)d_in[7];

    float* out = (float*)d_out;
    float* ws  = (float*)d_ws;

    float* avg = ws;                                   // 2*256*65536 floats
    float* Wp  = avg + (size_t)2 * 256 * NPLANE;       // 65536 floats (pair-packed)
    float* ssq = Wp + 65536;                           // 2*256*4 floats
    float* origin = out;                               // reuse output buffer

    compose_wt<<<dim3(256), dim3(256), 0, stream>>>(W1, W2, Wp);
    pass1<<<dim3(512), dim3(256), 0, stream>>>(cen, avg, ssq);
    gemm_wmma<<<dim3(4096, 2), dim3(256), 0, stream>>>(avg, Wp, b2, origin);
    pass2<<<dim3(512), dim3(256), 0, stream>>>(cen, origin, ssq,
                                               n1w, n1b, n2w, n2b, out);
}